// Layer_67946382623148
// MI455X (gfx1250) — compile-verified
//
#include <hip/hip_runtime.h>

typedef float v2f __attribute__((ext_vector_type(2)));
typedef float v8f __attribute__((ext_vector_type(8)));

#define N_IN   4096
#define N_OUT  4096
#define N_W    16                 // GRID_SIZE + SPLINE_ORDER
#define KTOT   (N_IN * N_W)       // 65536 flattened reduction length
#define SPLITK 32
#define KLEN   (KTOT / SPLITK)    // 2048 (divisible by 64)

// ---------------------------------------------------------------------------
// Kernel 1: x = tanh(inputs); closed-form uniform cubic B-spline basis.
// Writes Bflat[i*16 + k] (row-major, matches W's ik flattening).
// ---------------------------------------------------------------------------
__global__ void __launch_bounds__(256)
kan_basis(const float* __restrict__ in, float* __restrict__ Bflat) {
    const int i = blockIdx.x * blockDim.x + threadIdx.x;
    const float x = tanhf(in[i]);
    // t = (x + 1) / h, h = 2/13  ->  t = (x + 1) * 6.5
    const float t  = (x + 1.0f) * 6.5f;
    int c = (int)floorf(t);
    c = c < 0 ? 0 : (c > 12 ? 12 : c);
    const float u   = t - (float)c;
    const float omu = 1.0f - u;
    const float u2  = u * u;
    const float u3  = u2 * u;
    const float s   = 1.0f / 6.0f;
    const float b0  = omu * omu * omu * s;
    const float b1  = (3.0f * u3 - 6.0f * u2 + 4.0f) * s;
    const float b2  = (-3.0f * u3 + 3.0f * u2 + 3.0f * u + 1.0f) * s;
    const float b3  = u3 * s;

    float v[16];
#pragma unroll
    for (int k = 0; k < 16; ++k) {
        float r = 0.0f;
        r = (k == c    ) ? b0 : r;
        r = (k == c + 1) ? b1 : r;
        r = (k == c + 2) ? b2 : r;
        r = (k == c + 3) ? b3 : r;
        v[k] = r;
    }
    float4* dst = (float4*)(Bflat + (size_t)i * 16);
    dst[0] = make_float4(v[0],  v[1],  v[2],  v[3]);
    dst[1] = make_float4(v[4],  v[5],  v[6],  v[7]);
    dst[2] = make_float4(v[8],  v[9],  v[10], v[11]);
    dst[3] = make_float4(v[12], v[13], v[14], v[15]);
}

// ---------------------------------------------------------------------------
// Kernel 2: streaming GEMV via V_WMMA_F32_16X16X4_F32.
//   A (16x4)  = basis chunk replicated across all 16 rows
//   B (4x16)  = W tile: column n = output o0+n, rows = 4 reduction elements
//   D = A*B + C : every row of D holds the 16 partial dot products.
// One wave per workgroup; grid = (256 output tiles) x (SPLITK k-slices).
// Per trip: 32 independent b64 loads staged first (max bytes in flight),
// then 16 WMMAs on two alternating accumulators (relaxed XDL RAW chain).
// ---------------------------------------------------------------------------
__global__ void __launch_bounds__(32)
kan_gemv_wmma(const float* __restrict__ W, const float* __restrict__ Bflat,
              float* __restrict__ part) {
    const int tile   = blockIdx.x;      // 0..255
    const int kslice = blockIdx.y;      // 0..SPLITK-1
    const int lane   = threadIdx.x;     // 0..31
    const int o0     = tile * 16;
    const int row    = lane & 15;       // B-matrix column (output index)
    const int half   = lane >> 4;       // which K-pair this lane holds

    const int j0 = kslice * KLEN;
    const float* wp = W + (size_t)(o0 + row) * KTOT + j0 + 2 * half;
    const float* bp = Bflat + j0 + 2 * half;

    v8f acc0 = {0.f, 0.f, 0.f, 0.f, 0.f, 0.f, 0.f, 0.f};
    v8f acc1 = {0.f, 0.f, 0.f, 0.f, 0.f, 0.f, 0.f, 0.f};

    for (int it = 0; it < KLEN / 64; ++it) {
        __builtin_prefetch(wp + 4096, 0, 1);   // ~16KB ahead per row
        v2f a[16], b[16];
#pragma unroll
        for (int q = 0; q < 16; ++q) {
            a[q] = *(const v2f*)(bp + 4 * q);  // basis fragment (L0-resident)
            b[q] = *(const v2f*)(wp + 4 * q);  // W fragment, sequential per lane
        }
#pragma unroll
        for (int q = 0; q < 16; ++q) {
            if (q & 1)
                acc1 = __builtin_amdgcn_wmma_f32_16x16x4_f32(
                           false, a[q], false, b[q], (short)0, acc1, false, false);
            else
                acc0 = __builtin_amdgcn_wmma_f32_16x16x4_f32(
                           false, a[q], false, b[q], (short)0, acc0, false, false);
        }
        wp += 64;
        bp += 64;
    }

    // Row M=0 of D lives in element 0 of the accumulator, lanes 0..15 (N=lane).
    if (lane < 16) {
        part[(size_t)kslice * N_OUT + o0 + lane] = acc0[0] + acc1[0];
    }
}

// ---------------------------------------------------------------------------
// Kernel 3: deterministic split-K reduction + bias.
// ---------------------------------------------------------------------------
__global__ void __launch_bounds__(256)
kan_reduce(const float* __restrict__ part, const float* __restrict__ bias,
           float* __restrict__ out) {
    const int o = blockIdx.x * blockDim.x + threadIdx.x;
    float s = bias[o];
#pragma unroll
    for (int k = 0; k < SPLITK; ++k) s += part[(size_t)k * N_OUT + o];
    out[o] = s;
}

// ---------------------------------------------------------------------------
extern "C" void kernel_launch(void* const* d_in, const int* in_sizes, int n_in,
                              void* d_out, int out_size, void* d_ws, size_t ws_size,
                              hipStream_t stream) {
    (void)in_sizes; (void)n_in; (void)out_size; (void)ws_size;
    const float* inputs = (const float*)d_in[0];
    const float* W      = (const float*)d_in[1];
    const float* bias   = (const float*)d_in[2];
    float*       out    = (float*)d_out;

    float* Bflat = (float*)d_ws;        // KTOT floats
    float* part  = Bflat + KTOT;        // SPLITK * N_OUT floats

    kan_basis<<<N_IN / 256, 256, 0, stream>>>(inputs, Bflat);
    kan_gemv_wmma<<<dim3(N_OUT / 16, SPLITK), 32, 0, stream>>>(W, Bflat, part);
    kan_reduce<<<N_OUT / 256, 256, 0, stream>>>(part, bias, out);
}